// MinNormSolver_68624987456030
// MI455X (gfx1250) — compile-verified
//
#include <hip/hip_runtime.h>
#include <math.h>

typedef __attribute__((ext_vector_type(2))) float v2f;
typedef __attribute__((ext_vector_type(8))) float v8f;

#define N_TASKS   16
#define WPB       8            // waves per block (wave32 -> 256 threads)
#define TPB       (WPB * 32)
#define MAX_ITER  250
#define STOP_CRIT 1e-6f
#define EPS_C     1e-8f
#define T_EPS_C   1e-7f

// ---------------------------------------------------------------------------
// Kernel 1: per-block partial Gram tiles via V_WMMA_F32_16X16X4_F32.
//
// Each wave owns grid-strided chunks of 64 consecutive k. Since A and B are
// the SAME registers (Gram), any assignment of global k values to WMMA chunk
// positions is valid as long as each k appears exactly once across the WMMAs.
// So lane L = (row = L&15, half = L>>4) loads float4's at kb + 8j + 4*half:
//   pair (x,y) -> WMMA chain 0,  pair (z,w) -> WMMA chain 1.
// 8 b128 loads + 16 WMMAs per 64-k chunk, two independent accumulators.
// ---------------------------------------------------------------------------
__global__ __launch_bounds__(TPB) void gram_partial_kernel(
    const float* __restrict__ V, long long K, float* __restrict__ partials)
{
    const int lane = threadIdx.x & 31;
    const int wib  = threadIdx.x >> 5;
    const int row  = lane & 15;
    const int half = lane >> 4;

    const long long wave  = (long long)blockIdx.x * WPB + wib;
    const long long waves = (long long)gridDim.x * WPB;

    const float* __restrict__ rowPtr = V + (long long)row * K;

    v8f acc0 = (v8f){0.f, 0.f, 0.f, 0.f, 0.f, 0.f, 0.f, 0.f};
    v8f acc1 = (v8f){0.f, 0.f, 0.f, 0.f, 0.f, 0.f, 0.f, 0.f};

    const long long nChunks = K >> 6;          // 64 k per chunk
    for (long long c = wave; c < nChunks; c += waves) {
        const float* __restrict__ p = rowPtr + (c << 6) + 4 * half;
        float4 t[8];
        #pragma unroll
        for (int j = 0; j < 8; ++j) {
            t[j] = *(const float4*)(p + 8 * j);   // global_load_b128
        }
        #pragma unroll
        for (int j = 0; j < 8; ++j) {
            v2f a0; a0.x = t[j].x; a0.y = t[j].y;
            v2f a1; a1.x = t[j].z; a1.y = t[j].w;
            acc0 = __builtin_amdgcn_wmma_f32_16x16x4_f32(
                false, a0, false, a0, (short)0, acc0, false, false);
            acc1 = __builtin_amdgcn_wmma_f32_16x16x4_f32(
                false, a1, false, a1, (short)0, acc1, false, false);
        }
    }

    // Tail: K % 64 leftovers, 4 k at a time with guarded loads (EXEC stays
    // full around the WMMA; guards lower to predicated loads + cndmask).
    for (long long kb = (nChunks << 6) + (wave << 2); kb < K; kb += (waves << 2)) {
        const long long k0 = kb + 2 * half;
        v2f a;
        a.x = (k0     < K) ? rowPtr[k0]     : 0.f;
        a.y = (k0 + 1 < K) ? rowPtr[k0 + 1] : 0.f;
        acc0 = __builtin_amdgcn_wmma_f32_16x16x4_f32(
            false, a, false, a, (short)0, acc0, false, false);
    }

    v8f acc = acc0 + acc1;

    // Deterministic in-block reduction across the 8 waves via LDS.
    __shared__ float red[WPB][32][8];
    #pragma unroll
    for (int e = 0; e < 8; ++e) red[wib][lane][e] = acc[e];
    __syncthreads();

    const int t  = threadIdx.x;      // TPB == 256, one slot per thread
    const int e  = t >> 5;
    const int ln = t & 31;
    float s = 0.f;
    #pragma unroll
    for (int w = 0; w < WPB; ++w) s += red[w][ln][e];
    partials[(long long)blockIdx.x * 256 + t] = s;
}

// ---------------------------------------------------------------------------
// Kernel 2: deterministic cross-block reduction + un-permute WMMA C/D layout
// (VGPR e, lanes 0-15 -> M=e,N=lane; lanes 16-31 -> M=e+8,N=lane-16) into G.
// ---------------------------------------------------------------------------
__global__ __launch_bounds__(256) void gram_reduce_kernel(
    const float* __restrict__ partials, int nblocks, float* __restrict__ G)
{
    const int t = threadIdx.x;
    float s = 0.f;
    for (int b = 0; b < nblocks; ++b)
        s += partials[(long long)b * 256 + t];
    const int e    = t >> 5;
    const int lane = t & 31;
    const int m    = e + ((lane >> 4) << 3);
    const int n    = lane & 15;
    G[m * 16 + n] = s;
}

// ---------------------------------------------------------------------------
// Kernel 3: Frank-Wolfe min-norm solver on the 16x16 Gram matrix.
// Single thread; faithfully mirrors the JAX reference semantics.
// ---------------------------------------------------------------------------
__device__ __forceinline__ void line_solver(float v11, float v12, float v22,
                                            float& gamma, float& cost)
{
    const float gamma0 = (v22 - v12) / (v11 + v22 - 2.0f * v12 + EPS_C);
    const float cost0  = v22 + gamma0 * (v12 - v22);
    if (v12 >= v11)      { gamma = 1.0f;   cost = v11;   }
    else if (v12 >= v22) { gamma = 0.0f;   cost = v22;   }
    else                 { gamma = gamma0; cost = cost0; }
}

__device__ __forceinline__ void proj_simplex(const float* g, float* out)
{
    float s[N_TASKS];
    for (int i = 0; i < N_TASKS; ++i) s[i] = g[i];
    // insertion sort, descending
    for (int i = 1; i < N_TASKS; ++i) {
        const float key = s[i];
        int j = i - 1;
        while (j >= 0 && s[j] < key) { s[j + 1] = s[j]; --j; }
        s[j + 1] = key;
    }
    float tmax[N_TASKS];
    float cs = 0.f;
    for (int i = 0; i < N_TASKS; ++i) {
        cs += s[i];
        tmax[i] = (cs - 1.0f) / (float)(i + 1);
    }
    int idx = -1;
    for (int i = 0; i < N_TASKS - 1; ++i) {
        if (tmax[i] > s[i + 1]) { idx = i; break; }   // first True
    }
    const float tf = (idx >= 0) ? tmax[idx] : tmax[N_TASKS - 1];
    for (int i = 0; i < N_TASKS; ++i)
        out[i] = fmaxf(g[i] - tf, 0.0f);
}

__device__ __forceinline__ void next_point(const float* cur, const float* grad,
                                           float* out)
{
    float sum = 0.f;
    for (int k = 0; k < N_TASKS; ++k) sum += grad[k];
    float pg[N_TASKS];
    for (int k = 0; k < N_TASKS; ++k) pg[k] = grad[k] - sum / (float)N_TASKS;

    bool any1 = false, any2 = false;
    float min1 = INFINITY, min2 = INFINITY;
    for (int k = 0; k < N_TASKS; ++k) {
        const float pgs = (pg[k] == 0.0f) ? 1.0f : pg[k];
        const float tm1 = -cur[k] / pgs;
        const float tm2 = (1.0f - cur[k]) / pgs;
        if (pg[k] < 0.0f && tm1 > T_EPS_C) { any1 = true; min1 = fminf(min1, tm1); }
        if (pg[k] > 0.0f && tm2 > T_EPS_C) { any2 = true; min2 = fminf(min2, tm2); }
    }
    float t = any1 ? min1 : 1.0f;
    if (any2) t = fminf(t, min2);

    float g[N_TASKS];
    for (int k = 0; k < N_TASKS; ++k) g[k] = pg[k] * t + cur[k];
    proj_simplex(g, out);
}

__global__ void fw_solve_kernel(const float* __restrict__ Gin,
                                float* __restrict__ out)
{
    if (threadIdx.x != 0 || blockIdx.x != 0) return;

    float G[N_TASKS][N_TASKS];
    for (int i = 0; i < N_TASKS; ++i)
        for (int j = 0; j < N_TASKS; ++j)
            G[i][j] = Gin[i * N_TASKS + j];

    // planar init: best pairwise min-norm point (first argmin, triu row-major)
    float best_cost = INFINITY, bgamma = 0.f;
    int bi = 0, bj = 1;
    for (int i = 0; i < N_TASKS; ++i) {
        for (int j = i + 1; j < N_TASKS; ++j) {
            float gamma, cost;
            line_solver(G[i][i], G[i][j], G[j][j], gamma, cost);
            if (cost < best_cost) { best_cost = cost; bi = i; bj = j; bgamma = gamma; }
        }
    }
    float s[N_TASKS];
    for (int k = 0; k < N_TASKS; ++k) s[k] = 0.f;
    s[bi] = bgamma;
    s[bj] = 1.0f - bgamma;

    for (int it = 0; it < MAX_ITER; ++it) {
        float Gs[N_TASKS], grad[N_TASKS];
        for (int i = 0; i < N_TASKS; ++i) {
            float acc = 0.f;
            for (int j = 0; j < N_TASKS; ++j) acc += G[i][j] * s[j];
            Gs[i] = acc;
            grad[i] = -acc;
        }
        float npt[N_TASKS];
        next_point(s, grad, npt);

        float Gn[N_TASKS];
        for (int i = 0; i < N_TASKS; ++i) {
            float acc = 0.f;
            for (int j = 0; j < N_TASKS; ++j) acc += G[i][j] * npt[j];
            Gn[i] = acc;
        }
        float v11 = 0.f, v12 = 0.f, v22 = 0.f;
        for (int k = 0; k < N_TASKS; ++k) {
            v11 += s[k] * Gs[k];
            v12 += s[k] * Gn[k];
            v22 += npt[k] * Gn[k];
        }
        float gam, cst;
        line_solver(v11, v12, v22, gam, cst);

        float ns[N_TASKS], diff = 0.f;
        for (int k = 0; k < N_TASKS; ++k) {
            ns[k] = gam * s[k] + (1.0f - gam) * npt[k];
            diff += fabsf(ns[k] - s[k]);
        }
        if (diff < STOP_CRIT) break;       // freeze: output PRE-update s
        for (int k = 0; k < N_TASKS; ++k) s[k] = ns[k];
    }

    for (int k = 0; k < N_TASKS; ++k) out[k] = s[k];
}

// ---------------------------------------------------------------------------
extern "C" void kernel_launch(void* const* d_in, const int* in_sizes, int n_in,
                              void* d_out, int out_size, void* d_ws, size_t ws_size,
                              hipStream_t stream)
{
    const float* V = (const float*)d_in[0];
    const long long total = (long long)in_sizes[0];
    const long long K = total / N_TASKS;   // D_FEAT

    float* G        = (float*)d_ws;        // 256 floats
    float* partials = G + 256;             // nblocks * 256 floats

    int nblocks = 512;
    const long long maxBlocks = ((long long)(ws_size / 4) - 256) / 256;
    if (maxBlocks < (long long)nblocks) nblocks = (int)maxBlocks;
    if (nblocks < 1) nblocks = 1;

    gram_partial_kernel<<<nblocks, TPB, 0, stream>>>(V, K, partials);
    gram_reduce_kernel<<<1, 256, 0, stream>>>(partials, nblocks, G);
    fw_solve_kernel<<<1, 32, 0, stream>>>(G, (float*)d_out);
}